// GraphNet_88596585382209
// MI455X (gfx1250) — compile-verified
//
#include <hip/hip_runtime.h>

// ---------------------------------------------------------------------------
// GraphNet message-passing on MI455X (gfx1250, wave32, WMMA).
// N=128 nodes, B=8 batch, F=512 feat. All fp32; GEMMs use V_WMMA_F32_16X16X4_F32.
// ---------------------------------------------------------------------------

#define N_NODES 128
#define BATCH   8
#define FDIM    512
#define MROWS   (N_NODES * BATCH)     // 1024 rows (m = i*BATCH + b)
#define NEG_SLOPE 0.1f

typedef __attribute__((ext_vector_type(2))) float v2f;
typedef __attribute__((ext_vector_type(8))) float v8f;

// ---------------------------------------------------------------------------
// Kernel 1: virtual GEMM  C[1024 x 1024] = nodes[1024 x 512] @ [W3a | W3b]
//   columns   0..511  -> Ai = nodes @ W3[:F]            (stored to Ai)
//   columns 512..1023 -> Cj = nodes @ W3[F:] + b3       (stored to Cj)
// One wave computes one 16x16 tile; K loop in steps of 4 via f32 WMMA.
// ---------------------------------------------------------------------------
__global__ __launch_bounds__(256)
void gemm_w3_kernel(const float* __restrict__ nodes,  // [1024, 512]
                    const float* __restrict__ W3,     // [1024, 512] (two stacked 512x512)
                    const float* __restrict__ b3,     // [512]
                    float* __restrict__ Ai,           // [1024, 512]
                    float* __restrict__ Cj)           // [1024, 512]
{
    const int wave = (blockIdx.x * blockDim.x + threadIdx.x) >> 5;  // uniform per wave
    const int lane = threadIdx.x & 31;
    const int TN = (2 * FDIM) / 16;          // 64 tiles along virtual N
    const int tm = wave / TN;                // 0..127 (M has only 64 tiles)
    const int tn = wave % TN;                // 0..63
    const int m0 = tm * 16;
    const int n0 = tn * 16;

    const bool second = (n0 >= FDIM);                      // wave-uniform
    const float* __restrict__ Wb = second ? (W3 + FDIM * FDIM) : W3;
    const int nc = n0 & (FDIM - 1);

    // A frag: lane<16 holds M=lane, K={0,1}; lane>=16 holds M=lane-16, K={2,3}
    const int rc = lane & 15;          // row (A) / col (B)
    const int kb = (lane >> 4) << 1;   // 0 or 2

    v8f acc = {};
    const float* __restrict__ ap = nodes + (m0 + rc) * FDIM + kb;
    const float* __restrict__ bp = Wb + kb * FDIM + nc + rc;

    for (int k = 0; k < FDIM; k += 4) {
        v2f a, b;
        a.x = ap[k + 0];
        a.y = ap[k + 1];
        b.x = bp[k * FDIM];            // B[kb+0][col]
        b.y = bp[(k + 1) * FDIM];      // B[kb+1][col]
        acc = __builtin_amdgcn_wmma_f32_16x16x4_f32(
            /*neg_a=*/false, a, /*neg_b=*/false, b,
            /*c_mod=*/(short)0, acc, /*reuse_a=*/false, /*reuse_b=*/false);
    }

    // C/D layout: VGPR r -> row m0 + (lane>=16 ? 8 : 0) + r, col = lane&15
    float* __restrict__ dst = second ? Cj : Ai;
    const float bias = second ? b3[nc + rc] : 0.0f;
    const int mrow = m0 + ((lane >> 4) << 3);
#pragma unroll
    for (int r = 0; r < 8; ++r) {
        dst[(mrow + r) * FDIM + nc + rc] = acc[r] + bias;
    }
}

// ---------------------------------------------------------------------------
// Kernel 2: pairwise message reduction.
//   lrelu(z) = a*z + d*|z|,  a=(1+s)/2, d=(1-s)/2
//   sum_j lrelu(X_i + C_j) = a*(N*X_i + sum_j C_j) + d*sum_j |X_i + C_j|
//   message = (sum - lrelu(X_i + C_i)) / (N-1)
// Block = 256 threads, handles one batch b and a 32-feature tile.
// LDS panels: X[128x32], C[128x32] (32 KB). Lane f reads C[j][f] (bank-clean
// broadcast across i-groups); 16 accumulators per thread.
// ---------------------------------------------------------------------------
__global__ __launch_bounds__(256)
void pairwise_kernel(const float* __restrict__ Ai,   // [1024, 512]
                     const float* __restrict__ Cj,   // [1024, 512] (incl. b3)
                     float* __restrict__ Msg)        // [1024, 512]
{
    __shared__ float Xs[N_NODES * 32];
    __shared__ float Cs[N_NODES * 32];

    const int b  = blockIdx.y;
    const int f0 = blockIdx.x * 32;
    const int t  = threadIdx.x;

    for (int idx = t; idx < N_NODES * 32; idx += 256) {
        const int i = idx >> 5;
        const int f = idx & 31;
        const int g = (i * BATCH + b) * FDIM + f0 + f;
        Xs[idx] = Ai[g];
        Cs[idx] = Cj[g];
    }
    __syncthreads();

    const int f  = t & 31;     // feature within tile
    const int ig = t >> 5;     // i-group 0..7; thread owns i = ig + 8*ii

    float x[16], acc[16];
    float csum = 0.0f;
#pragma unroll
    for (int ii = 0; ii < 16; ++ii) {
        x[ii]   = Xs[(ig + 8 * ii) * 32 + f];
        acc[ii] = 0.0f;
    }

    for (int j = 0; j < N_NODES; ++j) {
        const float c = Cs[j * 32 + f];
        csum += c;
#pragma unroll
        for (int ii = 0; ii < 16; ++ii)
            acc[ii] += fabsf(x[ii] + c);
    }

    const float ca = 0.5f * (1.0f + NEG_SLOPE);
    const float cd = 0.5f * (1.0f - NEG_SLOPE);
    const float inv = 1.0f / (float)(N_NODES - 1);
#pragma unroll
    for (int ii = 0; ii < 16; ++ii) {
        const int i = ig + 8 * ii;
        const float zc   = x[ii] + Cs[i * 32 + f];          // diag j == i
        const float diag = (zc >= 0.0f) ? zc : NEG_SLOPE * zc;
        const float s    = ca * ((float)N_NODES * x[ii] + csum) + cd * acc[ii];
        Msg[(i * BATCH + b) * FDIM + f0 + f] = (s - diag) * inv;
    }
}

// ---------------------------------------------------------------------------
// Kernel 3: out = lrelu(Msg @ W5 + b5) + nodes   (fused epilogue)
// Same per-wave 16x16 WMMA tiling as kernel 1.
// ---------------------------------------------------------------------------
__global__ __launch_bounds__(256)
void gemm_w5_kernel(const float* __restrict__ Msg,   // [1024, 512]
                    const float* __restrict__ W5,    // [512, 512]
                    const float* __restrict__ b5,    // [512]
                    const float* __restrict__ nodes, // [1024, 512]
                    float* __restrict__ out)         // [1024, 512]
{
    const int wave = (blockIdx.x * blockDim.x + threadIdx.x) >> 5;
    const int lane = threadIdx.x & 31;
    const int TN = FDIM / 16;                // 32
    const int tm = wave / TN;                // 0..63
    const int tn = wave % TN;
    const int m0 = tm * 16;
    const int n0 = tn * 16;

    const int rc = lane & 15;
    const int kb = (lane >> 4) << 1;

    v8f acc = {};
    const float* __restrict__ ap = Msg + (m0 + rc) * FDIM + kb;
    const float* __restrict__ bp = W5 + kb * FDIM + n0 + rc;

    for (int k = 0; k < FDIM; k += 4) {
        v2f a, b;
        a.x = ap[k + 0];
        a.y = ap[k + 1];
        b.x = bp[k * FDIM];
        b.y = bp[(k + 1) * FDIM];
        acc = __builtin_amdgcn_wmma_f32_16x16x4_f32(
            false, a, false, b, (short)0, acc, false, false);
    }

    const float bias = b5[n0 + rc];
    const int mrow = m0 + ((lane >> 4) << 3);
#pragma unroll
    for (int r = 0; r < 8; ++r) {
        const int m = mrow + r;
        float z = acc[r] + bias;
        z = (z >= 0.0f) ? z : NEG_SLOPE * z;
        out[m * FDIM + n0 + rc] = z + nodes[m * FDIM + n0 + rc];
    }
}

// ---------------------------------------------------------------------------
// Launcher. Inputs (setup_inputs order): nodes, W3, b3, W5, b5.
// Workspace: Ai (2MB) | Cj (2MB) | Msg (2MB).
// ---------------------------------------------------------------------------
extern "C" void kernel_launch(void* const* d_in, const int* in_sizes, int n_in,
                              void* d_out, int out_size, void* d_ws, size_t ws_size,
                              hipStream_t stream) {
    const float* nodes = (const float*)d_in[0];
    const float* W3    = (const float*)d_in[1];
    const float* b3    = (const float*)d_in[2];
    const float* W5    = (const float*)d_in[3];
    const float* b5    = (const float*)d_in[4];
    float* out = (float*)d_out;

    float* Ai  = (float*)d_ws;
    float* Cj  = Ai + MROWS * FDIM;
    float* Msg = Cj + MROWS * FDIM;

    // GEMM-W3: 64 x 64 tiles, grid of 8192 waves; 8 waves (256 threads) per block.
    gemm_w3_kernel<<<(64 * 128) / 8, 256, 0, stream>>>(nodes, W3, b3, Ai, Cj);

    // Pairwise: grid (F/32, B) = (16, 8) blocks of 256 threads.
    pairwise_kernel<<<dim3(FDIM / 32, BATCH), 256, 0, stream>>>(Ai, Cj, Msg);

    // GEMM-W5: 64 x 32 tiles = 2048 waves -> 256 blocks.
    gemm_w5_kernel<<<(64 * 32) / 8, 256, 0, stream>>>(Msg, W5, b5, nodes, out);
}